// RecurrentCrossLinearAttention_17171279249470
// MI455X (gfx1250) — compile-verified
//
#include <hip/hip_runtime.h>
#include <cstdint>
#include <cstddef>

// Problem constants (from reference): B=8, S=8192, N=16, D=64, M=64
#define B_      8
#define S_      8192
#define N_      16
#define D_      64
#define M_      64
#define EPS_    1e-6f

#define NCHUNK  16                    // S-chunks per (b,n)  -> 2048 WGs in pass 1
#define SCHUNK  (S_ / NCHUNK)         // 512 sequence positions per chunk
#define TS      32                    // LDS tile height (sequence positions)
#define NT      (SCHUNK / TS)         // 16 tiles per chunk
#define PART_STRIDE (M_ * D_ + D_)    // 4096 Sst + 64 Z floats per partial

#define LSTRIDE (D_ + 16)             // LDS row stride (pad 16 -> lane halves hit disjoint banks)

// Output layout: V [B,N,M] | Sst [B,N,M,D] | Z [B,N,D]
#define OUT_V_OFF 0
#define OUT_S_OFF (B_ * N_ * M_)                  // 8192
#define OUT_Z_OFF (OUT_S_OFF + B_ * N_ * M_ * D_) // 532480

typedef float v2f __attribute__((ext_vector_type(2)));
typedef float v8f __attribute__((ext_vector_type(8)));

__device__ __forceinline__ float phi_feat(float x) {
    // elu(x)+1 : x>0 -> x+1 ; x<=0 -> exp(x)
    return x > 0.0f ? x + 1.0f : __expf(x);
}

// Issue two async global->LDS b128 loads covering 8 consecutive floats.
__device__ __forceinline__ void async_v_load(uint32_t lds_off, const float* gp) {
    asm volatile(
        "global_load_async_to_lds_b128 %0, %2, off\n\t"
        "global_load_async_to_lds_b128 %1, %2, off offset:16"
        :: "v"(lds_off), "v"(lds_off + 16u), "v"(gp)
        : "memory");
}

// ---------------------------------------------------------------------------
// Pass 1: per (b, n, chunk) compute partial Sst[m,d] (via fp32 WMMA) and Z[d].
// Grid: B*N*NCHUNK blocks of 256 threads (8 wave32).
// Wave w owns output rows m0..m0+15 and cols d0..d0+31 (two 16x16 WMMA tiles).
// V tiles are streamed global->LDS with async-to-LDS loads, double buffered
// one S-tile ahead (ASYNCcnt-managed); K goes through VGPRs for elu+1 & mask.
// ---------------------------------------------------------------------------
__global__ __launch_bounds__(256)
void la_pass1(const float* __restrict__ key,
              const float* __restrict__ value,
              const unsigned char* __restrict__ key_mask,
              float* __restrict__ ws)
{
    __shared__ float Kt[TS][LSTRIDE];       // feature-mapped, masked keys
    __shared__ float Vb[2][TS][LSTRIDE];    // raw values, double buffered

    const int t     = threadIdx.x;
    const int chunk = blockIdx.x % NCHUNK;
    const int bn    = blockIdx.x / NCHUNK;     // 0..127
    const int n     = bn % N_;
    const int b     = bn / N_;

    const int lane = t & 31;
    const int wave = t >> 5;                   // 0..7
    const int m0   = (wave & 3) * 16;          // output row tile
    const int d0   = (wave >> 2) * 32;         // output col slab (2 tiles)
    const int half = lane >> 4;                // 0: K=0,1  1: K=2,3 (assumed f32 A/B layout)
    const int lm   = lane & 15;

    v8f c0 = {};                               // accumulator tile (m0.., d0..d0+15)
    v8f c1 = {};                               // accumulator tile (m0.., d0+16..d0+31)

    const int ldrow = t >> 3;                  // 0..31 : sequence row within tile
    const int ldcol = (t & 7) * 8;             // 8 floats per thread per matrix

    const int s_base = chunk * SCHUNK;
    float zacc = 0.0f;                         // threads 0..63 accumulate Z[d=t]

    // Per-thread fixed LDS offsets of the two V buffers (low 32 bits of the
    // generic LDS pointer are the LDS byte address on gfx1250).
    const uint32_t vlds0 = (uint32_t)(uintptr_t)&Vb[0][ldrow][ldcol];
    const uint32_t vlds1 = (uint32_t)(uintptr_t)&Vb[1][ldrow][ldcol];

    // Per-thread base global pointer for V rows of this chunk.
    const float* vp0 = value +
        ((((size_t)b * S_ + (s_base + ldrow)) * N_ + n) * M_ + ldcol);
    const size_t vstep = (size_t)TS * N_ * M_;   // advance TS sequence rows

    // Prologue: async-load V tile 0 into buffer 0 (nobody has read it yet).
    async_v_load(vlds0, vp0);

    for (int tile = 0; tile < NT; ++tile) {
        const int cur = tile & 1;
        const int s_g = s_base + tile * TS + ldrow;
        const float* kp = key + ((((size_t)b * S_ + s_g) * N_ + n) * D_ + ldcol);
        const bool   mk = key_mask[(size_t)b * S_ + s_g] != 0;

        float4 k0 = ((const float4*)kp)[0];
        float4 k1 = ((const float4*)kp)[1];

        __syncthreads();   // barrier A: previous tile's Kt / Vb[cur^1] readers done

        // Async-prefetch V for tile+1 into the other buffer (now safe to write).
        if (tile + 1 < NT)
            async_v_load(cur ? vlds0 : vlds1, vp0 + (size_t)(tile + 1) * vstep);

        if (mk) {
            k0 = make_float4(0.f, 0.f, 0.f, 0.f);
            k1 = make_float4(0.f, 0.f, 0.f, 0.f);
        } else {
            k0.x = phi_feat(k0.x); k0.y = phi_feat(k0.y);
            k0.z = phi_feat(k0.z); k0.w = phi_feat(k0.w);
            k1.x = phi_feat(k1.x); k1.y = phi_feat(k1.y);
            k1.z = phi_feat(k1.z); k1.w = phi_feat(k1.w);
        }
        *(float4*)&Kt[ldrow][ldcol]     = k0;
        *(float4*)&Kt[ldrow][ldcol + 4] = k1;

        if (tile + 1 < NT) {   // warm L2 for next K tile
            __builtin_prefetch(kp + (size_t)TS * N_ * D_, 0, 3);
            // This wave's tile-(tile) V asyncs are the 2 oldest of 4 in
            // flight; async loads complete in order -> waiting <=2 retires them.
            asm volatile("s_wait_asynccnt 0x2" ::: "memory");
        } else {
            asm volatile("s_wait_asynccnt 0x0" ::: "memory");
        }

        __syncthreads();   // barrier B: Kt stores + V(tile) async writes visible

        // D[m,d] += A[m,k] * B[k,d] ; A = V^T fragment, B = K fragment.
        // Assumed f32 16x16x4 per-lane layout (symmetric A/B):
        //   v0: K = 0 (lanes 0-15) / K = 2 (lanes 16-31)
        //   v1: K = 1 (lanes 0-15) / K = 3 (lanes 16-31)
        const float (* __restrict__ Vt)[LSTRIDE] = Vb[cur];
#pragma unroll
        for (int sk = 0; sk < TS; sk += 4) {
            const int r0 = sk + 2 * half;
            v2f a, b0v, b1v;
            a.x   = Vt[r0][m0 + lm];        a.y   = Vt[r0 + 1][m0 + lm];
            b0v.x = Kt[r0][d0 + lm];        b0v.y = Kt[r0 + 1][d0 + lm];
            b1v.x = Kt[r0][d0 + 16 + lm];   b1v.y = Kt[r0 + 1][d0 + 16 + lm];
            c0 = __builtin_amdgcn_wmma_f32_16x16x4_f32(
                     false, a, false, b0v, (short)0, c0, false, false);
            c1 = __builtin_amdgcn_wmma_f32_16x16x4_f32(
                     false, a, false, b1v, (short)0, c1, false, false);
        }

        // Z[d] partial: fixed-order column sum by threads 0..63 (deterministic)
        if (t < D_) {
#pragma unroll
            for (int s = 0; s < TS; ++s) zacc += Kt[s][t];
        }
    }

    // Spill partials: C/D layout -> lane l<16: vgpr r = D[m0+r, d0+l];
    //                 lane l>=16: vgpr r = D[m0+8+r, d0+l-16]
    float* part = ws + ((size_t)bn * NCHUNK + chunk) * PART_STRIDE;
    const int rowb = m0 + half * 8;
#pragma unroll
    for (int r = 0; r < 8; ++r) {
        part[(rowb + r) * D_ + d0 + lm]      = c0[r];
        part[(rowb + r) * D_ + d0 + 16 + lm] = c1[r];
    }
    if (t < D_) part[M_ * D_ + t] = zacc;
}

// ---------------------------------------------------------------------------
// Pass 2: per (b,n) reduce NCHUNK partials (fixed order), write Sst and Z,
// then compute Qf = elu(q)+1, qz = 1/(Qf.Z + eps), V_out[m] = qz * Qf.Sst[m,:]
// ---------------------------------------------------------------------------
__global__ __launch_bounds__(256)
void la_pass2(const float* __restrict__ query,
              const float* __restrict__ ws,
              float* __restrict__ out)
{
    __shared__ float Ssum[M_ * D_];
    __shared__ float Zs[D_];
    __shared__ float Qf[D_];
    __shared__ float qz_s;

    const int t  = threadIdx.x;
    const int bn = blockIdx.x;    // 0..127

    const float* pbase = ws + (size_t)bn * NCHUNK * PART_STRIDE;

#pragma unroll
    for (int i = 0; i < 16; ++i) {
        const int e = t + i * 256;            // coalesced across threads
        float acc = 0.0f;
        for (int c = 0; c < NCHUNK; ++c)
            acc += pbase[(size_t)c * PART_STRIDE + e];
        Ssum[e] = acc;
        out[OUT_S_OFF + (size_t)bn * (M_ * D_) + e] = acc;
    }
    if (t < D_) {
        float z = 0.0f;
        for (int c = 0; c < NCHUNK; ++c)
            z += pbase[(size_t)c * PART_STRIDE + M_ * D_ + t];
        Zs[t] = z;
        out[OUT_Z_OFF + (size_t)bn * D_ + t] = z;
        Qf[t] = phi_feat(query[(size_t)bn * D_ + t]);
    }
    __syncthreads();
    if (t == 0) {
        float dot = 0.0f;
        for (int d = 0; d < D_; ++d) dot += Qf[d] * Zs[d];
        qz_s = 1.0f / (dot + EPS_);
    }
    __syncthreads();
    if (t < M_) {
        float acc = 0.0f;
        for (int d = 0; d < D_; ++d) acc += Qf[d] * Ssum[t * D_ + d];
        out[OUT_V_OFF + (size_t)bn * M_ + t] = acc * qz_s;
    }
}

// ---------------------------------------------------------------------------
extern "C" void kernel_launch(void* const* d_in, const int* in_sizes, int n_in,
                              void* d_out, int out_size, void* d_ws, size_t ws_size,
                              hipStream_t stream)
{
    (void)in_sizes; (void)n_in; (void)out_size; (void)ws_size;

    const float*         query    = (const float*)d_in[0];         // [B,N,D]
    const float*         key      = (const float*)d_in[1];         // [B,S,N,D]
    const float*         value    = (const float*)d_in[2];         // [B,S,N,M]
    const unsigned char* key_mask = (const unsigned char*)d_in[3]; // [B,S] bool

    float* out = (float*)d_out;
    float* ws  = (float*)d_ws;   // needs B*N*NCHUNK*PART_STRIDE*4 = ~32.5 MB

    la_pass1<<<dim3(B_ * N_ * NCHUNK), dim3(256), 0, stream>>>(key, value, key_mask, ws);
    la_pass2<<<dim3(B_ * N_), dim3(256), 0, stream>>>(query, ws, out);
}